// RTree_9328668967711
// MI455X (gfx1250) — compile-verified
//
#include <hip/hip_runtime.h>
#include <cstdint>

// Problem constants (B, R, C) = (8, 1024, 1024); EXTENTS give cell size 80/1024.
constexpr int   kB     = 8;
constexpr int   kR     = 1024;
constexpr int   kC     = 1024;
constexpr int   kPlane = kR * kC;        // 1 << 20
constexpr int   kN     = kB * kPlane;    // 1 << 23  (8M pixels)
constexpr float kGS    = 0.078125f;      // (40-(-40))/1024, exact in f32
constexpr float kThr   = 0.05f;

// ---------------------------------------------------------------------------
// Kernel 1: zero the 128MB accumulator (b128 stores; it will live in L2).
// ---------------------------------------------------------------------------
__global__ __launch_bounds__(256) void rtree_zero(float4* __restrict__ out) {
  const int i = blockIdx.x * 256 + threadIdx.x;   // grid sized exactly: i < kN
  out[i] = make_float4(0.f, 0.f, 0.f, 0.f);
}

// ---------------------------------------------------------------------------
// Kernel 2: masked scatter-add of [1, conf, vx, vy] into displaced cells.
// 4 cells per thread -> every input stream is a b128 load (512B per wave per
// request), which is what it takes to keep 23.3 TB/s of HBM fed. The 128MB
// accumulator fits in the 192MB L2, so the ~30M global_atomic_add_f32 resolve
// at L2. pixel_pred is staged via the gfx1250 async global->LDS path.
// ---------------------------------------------------------------------------
__global__ __launch_bounds__(256) void rtree_scatter(
    const int*   __restrict__ vox,
    const float* __restrict__ pix,
    const float* __restrict__ conf,
    const float* __restrict__ off,
    const float* __restrict__ vel,
    float*       __restrict__ out)
{
  const int t = blockIdx.x * 256 + threadIdx.x;   // t < kN/4
  const int i = t << 2;                           // first of 4 contiguous cells

  // --- CDNA5 async global->LDS staging of the pixel stream (ASYNCcnt path) ---
  __shared__ float4 s_pix[256];
  {
    const uint32_t laddr = (uint32_t)(uintptr_t)(&s_pix[threadIdx.x]); // low 32 bits = LDS byte addr
    const uint64_t gaddr = (uint64_t)(uintptr_t)(pix + i);
    asm volatile("global_load_async_to_lds_b128 %0, %1, off"
                 :: "v"(laddr), "v"(gaddr) : "memory");
  }

  const int b  = i >> 20;
  const int rc = i & (kPlane - 1);
  const int r  = rc >> 10;
  const int c  = rc & (kC - 1);

  // Issue all remaining loads up front as b128 (latency hiding; memory-bound).
  const int    pbase = b * (2 * kPlane) + rc;
  const int4   vx4 = *(const int4*)  (vox  + i);
  const float4 o0g = *(const float4*)(off  + pbase);
  const float4 o1g = *(const float4*)(off  + pbase + kPlane);
  const float4 cfg = *(const float4*)(conf + i);
  const float4 w0g = *(const float4*)(vel  + pbase);
  const float4 w1g = *(const float4*)(vel  + pbase + kPlane);

  asm volatile("s_wait_asynccnt 0x0" ::: "memory");
  const float4 p4 = s_pix[threadIdx.x];

  const float pj[4] = {p4.x,  p4.y,  p4.z,  p4.w};
  const int   vj[4] = {vx4.x, vx4.y, vx4.z, vx4.w};
  const float aj[4] = {o0g.x, o0g.y, o0g.z, o0g.w};   // row offsets
  const float bj[4] = {o1g.x, o1g.y, o1g.z, o1g.w};   // col offsets
  const float cj[4] = {cfg.x, cfg.y, cfg.z, cfg.w};   // confidence
  const float xj[4] = {w0g.x, w0g.y, w0g.z, w0g.w};   // velocity x
  const float yj[4] = {w1g.x, w1g.y, w1g.z, w1g.w};   // velocity y

#pragma unroll
  for (int j = 0; j < 4; ++j) {
    // fg mask; skipping background cells is bit-exact (reference adds 0.0
    // there, and any count==0 cell is overwritten by -0.1 in finalize).
    if (!((pj[j] > kThr) && (vj[j] >= 0))) continue;

    // jnp.round = round-half-even; __float2int_rn = v_cvt_i32_f32 (RNE).
    // Keep the true division (not *1/kGS) to match reference rounding.
    int tr = r       + __float2int_rn(aj[j] / kGS);
    int tc = (c + j) + __float2int_rn(bj[j] / kGS);
    tr = min(max(tr, 0), kR - 1);
    tc = min(max(tc, 0), kC - 1);

    float* dst = out + ((((size_t)b << 20) + ((size_t)tr << 10) + (size_t)tc) << 2);
    unsafeAtomicAdd(dst + 0, 1.0f);   // count
    unsafeAtomicAdd(dst + 1, cj[j]);  // confidence
    unsafeAtomicAdd(dst + 2, xj[j]);  // velocity x
    unsafeAtomicAdd(dst + 3, yj[j]);  // velocity y
  }
}

// ---------------------------------------------------------------------------
// Kernel 3: where(count > 0, agg, -0.1). One b128 load per cell (L2-hit);
// empty cells (~40% for Poisson lambda~1) get a single b128 store of -0.1.
// ---------------------------------------------------------------------------
__global__ __launch_bounds__(256) void rtree_finalize(float4* __restrict__ out) {
  const int i = blockIdx.x * 256 + threadIdx.x;   // cell id, i < kN exactly
  const float4 v = out[i];
  if (!(v.x > 0.f)) {
    out[i] = make_float4(-0.1f, -0.1f, -0.1f, -0.1f);
  }
}

// ---------------------------------------------------------------------------
// Host entry
// ---------------------------------------------------------------------------
extern "C" void kernel_launch(void* const* d_in, const int* in_sizes, int n_in,
                              void* d_out, int out_size, void* d_ws, size_t ws_size,
                              hipStream_t stream) {
  const int*   vox  = (const int*)  d_in[0];  // voxel_count_gt (B,R,C) i32
  const float* pix  = (const float*)d_in[1];  // pixel_pred     (B,R,C) f32
  const float* conf = (const float*)d_in[2];  // confidence_pred(B,R,C) f32
  const float* off  = (const float*)d_in[3];  // offset_pred    (B,2,R,C) f32
  /* d_in[4] = view_index (B,R,C,5) i32 -- unused by the reference math:
     skipping it saves 160 MB of HBM reads. */
  const float* vel  = (const float*)d_in[5];  // velocity_pred  (B,2,R,C) f32
  float* out = (float*)d_out;                 // (B,R,C,4) f32

  const dim3 blk(256);

  rtree_zero    <<<dim3(kN / 256),       blk, 0, stream>>>((float4*)out);
  rtree_scatter <<<dim3(kN / (4 * 256)), blk, 0, stream>>>(vox, pix, conf, off, vel, out);
  rtree_finalize<<<dim3(kN / 256),       blk, 0, stream>>>((float4*)out);
}